// TrustRegionInverseCompositional_41901700939953
// MI455X (gfx1250) — compile-verified
//
#include <hip/hip_runtime.h>

#define B_      8
#define C_      8
#define H_      240
#define W_      320
#define HW_     (H_ * W_)            // 76800
#define CHW_    (C_ * HW_)           // 614400
#define BPB     300                  // blocks per batch: 300*256 == HW_ (1 pixel/thread)
#define THREADS 256
#define HUBER_D 0.2f
#define DAMP_   1e-6f
#define OCC_T   0.1f

// d_ws float layout:
//   [0,   128) : T, 4x4 row-major per batch
//   [128, 640) : G, 8x8 Gram per batch (rows 0..6 x cols 0..6 used; col 6 = JtR0)
//   [640, 704) : jtr, 8 floats per batch (6 used)
#define WS_T    0
#define WS_G    128
#define WS_JTR  640

typedef __attribute__((ext_vector_type(2))) float v2f;
typedef __attribute__((ext_vector_type(8))) float v8f;

static __device__ __forceinline__ int imini(int a, int b) { return a < b ? a : b; }
static __device__ __forceinline__ int imaxi(int a, int b) { return a > b ? a : b; }

// ---------------------------------------------------------------- exp_se3 ---
__device__ inline void exp_se3_dev(const float xi[6], float T[16]) {
    const float wx = xi[0], wy = xi[1], wz = xi[2];
    const float vx = xi[3], vy = xi[4], vz = xi[5];
    const float th2 = wx * wx + wy * wy + wz * wz;
    const float th  = sqrtf(th2);
    const bool  sm  = th < 1e-6f;
    const float ths = sm ? 1.0f : th;
    const float sn  = sinf(ths), cs = cosf(ths);
    const float A  = sm ? (1.0f - th2 * (1.0f / 6.0f))          : sn / ths;
    const float Bc = sm ? (0.5f - th2 * (1.0f / 24.0f))         : (1.0f - cs) / (ths * ths);
    const float Cc = sm ? (1.0f / 6.0f - th2 * (1.0f / 120.0f)) : (ths - sn) / (ths * ths * ths);

    float Kx[9] = { 0.f, -wz,  wy,
                    wz,  0.f, -wx,
                   -wy,  wx,  0.f };
    float K2[9];
#pragma unroll
    for (int i = 0; i < 3; ++i)
#pragma unroll
        for (int j = 0; j < 3; ++j)
            K2[i * 3 + j] = Kx[i * 3 + 0] * Kx[0 * 3 + j]
                          + Kx[i * 3 + 1] * Kx[1 * 3 + j]
                          + Kx[i * 3 + 2] * Kx[2 * 3 + j];
    float R[9], V[9];
#pragma unroll
    for (int i = 0; i < 3; ++i)
#pragma unroll
        for (int j = 0; j < 3; ++j) {
            const float id = (i == j) ? 1.0f : 0.0f;
            R[i * 3 + j] = id + A  * Kx[i * 3 + j] + Bc * K2[i * 3 + j];
            V[i * 3 + j] = id + Bc * Kx[i * 3 + j] + Cc * K2[i * 3 + j];
        }
    const float tx = V[0] * vx + V[1] * vy + V[2] * vz;
    const float ty = V[3] * vx + V[4] * vy + V[5] * vz;
    const float tz = V[6] * vx + V[7] * vy + V[8] * vz;
    T[0]  = R[0]; T[1]  = R[1]; T[2]  = R[2]; T[3]  = tx;
    T[4]  = R[3]; T[5]  = R[4]; T[6]  = R[5]; T[7]  = ty;
    T[8]  = R[6]; T[9]  = R[7]; T[10] = R[8]; T[11] = tz;
    T[12] = 0.f;  T[13] = 0.f;  T[14] = 0.f;  T[15] = 1.f;
}

// ----------------------------------------------- per-pixel precomputation ---
struct Pix {
    float px, py, iD;        // normalized coords + inverse depth
    int   off[4];            // clamped bilinear tap offsets into an HxW plane
    float mw[4];             // tap weights with validity folded in
    int   oc, owp, owm, ohp, ohm;  // center + gradient neighbor offsets
    bool  inview;            // !occluded
};

__device__ inline void pix_setup(
    int p, float fx, float fy, float cx, float cy,
    const float* Tl,                      // 12 floats: rows 0..2 of T
    const float* __restrict__ iD0b,       // invD0 + b*HW
    const float* __restrict__ iD1b,       // invD1 + b*HW
    Pix& P)
{
    const int h = p / W_;
    const int w = p % W_;
    const float px = ((float)w - cx) / fx;
    const float py = ((float)h - cy) / fy;
    const float iD = iD0b[p];
    P.px = px; P.py = py; P.iD = iD;

    float X = Tl[0] * px + Tl[1] * py + Tl[2]  + Tl[3]  * iD;
    float Y = Tl[4] * px + Tl[5] * py + Tl[6]  + Tl[7]  * iD;
    float Z = Tl[8] * px + Tl[9] * py + Tl[10] + Tl[11] * iD;
    if (fabsf(Z) < 1e-8f) Z = 1e-8f;

    const float u = fx * X / Z + cx;
    const float v = fy * Y / Z + cy;
    const float invDw = iD / Z;

    float u0f = floorf(u), v0f = floorf(v);
    const float du = u - u0f, dv = v - v0f;
    u0f = fminf(fmaxf(u0f, -1e6f), 1e6f);
    v0f = fminf(fmaxf(v0f, -1e6f), 1e6f);
    const int u0 = (int)u0f, v0 = (int)v0f;

    const float wt4[4] = { (1.f - du) * (1.f - dv), du * (1.f - dv),
                           (1.f - du) * dv,         du * dv };
    float d1w = 0.f;
#pragma unroll
    for (int t = 0; t < 4; ++t) {
        const int yy = v0 + (t >> 1);
        const int xx = u0 + (t & 1);
        const bool valid = (xx >= 0) && (xx < W_) && (yy >= 0) && (yy < H_);
        const int yc = imini(imaxi(yy, 0), H_ - 1);
        const int xc = imini(imaxi(xx, 0), W_ - 1);
        P.off[t] = yc * W_ + xc;
        P.mw[t]  = valid ? wt4[t] : 0.f;
        d1w += iD1b[P.off[t]] * P.mw[t];
    }

    const bool inlier = invDw > (d1w - OCC_T);
    P.inview = inlier && (u > 0.f) && (u < (float)W_) &&
               (v > 0.f) && (v < (float)H_);

    P.oc  = h * W_ + w;
    P.owp = h * W_ + imini(w + 1, W_ - 1);
    P.owm = h * W_ + imaxi(w - 1, 0);
    P.ohp = imini(h + 1, H_ - 1) * W_ + w;
    P.ohm = imaxi(h - 1, 0) * W_ + w;
}

// Jr[0..5] = Jacobian row, Jr[6] = residual (0 if occluded)
__device__ inline void row_from_pix(
    const Pix& P, float fx, float fy,
    const float* __restrict__ x0c,        // x0 plane for this (b,c)
    const float* __restrict__ x1c,        // x1 plane for this (b,c)
    float Jr[7])
{
    const float x1w = x1c[P.off[0]] * P.mw[0] + x1c[P.off[1]] * P.mw[1]
                    + x1c[P.off[2]] * P.mw[2] + x1c[P.off[3]] * P.mw[3];
    const float x0v = x0c[P.oc];
    Jr[6] = P.inview ? (x1w - x0v) : 0.f;

    const float gx = 0.5f * (x0c[P.owp] - x0c[P.owm]);
    const float gy = 0.5f * (x0c[P.ohp] - x0c[P.ohm]);

    const float a  = fx * gx;
    const float bg = fy * gy;
    const float px = P.px, py = P.py, iD = P.iD;
    const float xy = px * py;
    Jr[0] = a * (-xy)            + bg * (-(1.f + py * py));
    Jr[1] = a * (1.f + px * px)  + bg * (xy);
    Jr[2] = a * (-py)            + bg * (px);
    Jr[3] = a * iD;
    Jr[4] = bg * iD;
    Jr[5] = a * (-iD * px)       + bg * (-iD * py);
}

// ------------------------------------------------------------------ k_init ---
__global__ void k_init(const float* __restrict__ pose, float* __restrict__ ws) {
    const int t = threadIdx.x;                    // 64 threads
    for (int i = t; i < 576; i += 64) ws[WS_G + i] = 0.f;  // zero G + jtr
    if (t < B_) {
        float xi[6];
#pragma unroll
        for (int k = 0; k < 6; ++k) xi[k] = pose[t * 6 + k];
        float T[16];
        exp_se3_dev(xi, T);
#pragma unroll
        for (int k = 0; k < 16; ++k) ws[WS_T + t * 16 + k] = T[k];
    }
}

// ------------------------------------------------------------------ k_gram ---
// One thread = one pixel; loop over 8 channels. Per channel a wave stages its
// 32 rows [J0..J5,res,wt] in LDS, then folds them into the 16x16 Gram with
// 8x v_wmma_f32_16x16x4_f32 (A = (w*m)^T chunk, B = m chunk, K=4 rows each).
__global__ __launch_bounds__(THREADS) void k_gram(
    const float* __restrict__ x0, const float* __restrict__ x1,
    const float* __restrict__ invD0, const float* __restrict__ invD1,
    const float* __restrict__ Kp, const float* __restrict__ Tw,
    float* __restrict__ G, float* __restrict__ wout)
{
    __shared__ float S[8][32 * 8];               // 8 waves x 32 rows x 8 floats
    const int b    = blockIdx.x / BPB;
    const int blk  = blockIdx.x % BPB;
    const int tid  = threadIdx.x;
    const int wave = tid >> 5;
    const int lane = tid & 31;
    const int half = lane >> 4;
    const int cL   = lane & 15;
    const int p    = blk * THREADS + tid;        // pixel in batch, < HW_

    const float fx = Kp[b * 4 + 0], fy = Kp[b * 4 + 1];
    const float cx = Kp[b * 4 + 2], cy = Kp[b * 4 + 3];
    float Tl[12];
#pragma unroll
    for (int k = 0; k < 12; ++k) Tl[k] = Tw[b * 16 + k];

    const float* x0b  = x0    + b * CHW_;
    const float* x1b  = x1    + b * CHW_;
    const float* iD0b = invD0 + b * HW_;
    const float* iD1b = invD1 + b * HW_;
    float* Sw = S[wave];

    Pix P;
    pix_setup(p, fx, fy, cx, cy, Tl, iD0b, iD1b, P);

    v8f acc = {};                                // Gram accumulator (16x16)
    for (int c = 0; c < C_; ++c) {
        float Jr[7];
        row_from_pix(P, fx, fy, x0b + c * HW_, x1b + c * HW_, Jr);

        const float ar = fabsf(Jr[6]);
        const float wt = (ar <= HUBER_D) ? 1.f : HUBER_D / fmaxf(ar, 1e-8f);
        wout[b * CHW_ + c * HW_ + p] = wt;

        // stage: row = lane, record = [J0..J5, res, wt]
#pragma unroll
        for (int k = 0; k < 7; ++k) Sw[lane * 8 + k] = Jr[k];
        Sw[lane * 8 + 7] = wt;
        __syncthreads();

        // 8 chunks x 4 rows. f32 WMMA fragment layout: lanes 0-15 carry
        // K={0,1} in vgpr{0,1}; lanes 16-31 carry K={2,3}.
#pragma unroll
        for (int ch = 0; ch < 8; ++ch) {
            const int r0 = ch * 4 + half * 2;
            const int r1 = r0 + 1;
            const float m0 = (cL < 7) ? Sw[r0 * 8 + cL] : 0.f;
            const float m1 = (cL < 7) ? Sw[r1 * 8 + cL] : 0.f;
            const float w0 = Sw[r0 * 8 + 7];
            const float w1 = Sw[r1 * 8 + 7];
            v2f av; av.x = w0 * m0; av.y = w1 * m1;
            v2f bv; bv.x = m0;      bv.y = m1;
            acc = __builtin_amdgcn_wmma_f32_16x16x4_f32(
                false, av, false, bv, (short)0, acc, false, false);
        }
        __syncthreads();
    }

    // D layout: lanes 0-15, vgpr r -> (M=r, N=lane). Need M,N < 7.
    if (lane < 7) {
        float* Gb = G + b * 64;
#pragma unroll
        for (int r = 0; r < 7; ++r) atomicAdd(&Gb[r * 8 + lane], acc[r]);
    }
}

// ------------------------------------------------------------------- k_jtr ---
__global__ __launch_bounds__(THREADS) void k_jtr(
    const float* __restrict__ x0, const float* __restrict__ x1,
    const float* __restrict__ invD0, const float* __restrict__ invD1,
    const float* __restrict__ Kp, const float* __restrict__ Tw,
    float* __restrict__ jtr, const float* __restrict__ weights)
{
    const int b   = blockIdx.x / BPB;
    const int blk = blockIdx.x % BPB;
    const int tid = threadIdx.x;
    const int p   = blk * THREADS + tid;

    const float fx = Kp[b * 4 + 0], fy = Kp[b * 4 + 1];
    const float cx = Kp[b * 4 + 2], cy = Kp[b * 4 + 3];
    float Tl[12];
#pragma unroll
    for (int k = 0; k < 12; ++k) Tl[k] = Tw[b * 16 + k];

    const float* x0b  = x0    + b * CHW_;
    const float* x1b  = x1    + b * CHW_;
    const float* iD0b = invD0 + b * HW_;
    const float* iD1b = invD1 + b * HW_;

    Pix P;
    pix_setup(p, fx, fy, cx, cy, Tl, iD0b, iD1b, P);

    float acc6[6] = {0.f, 0.f, 0.f, 0.f, 0.f, 0.f};
#pragma unroll
    for (int c = 0; c < C_; ++c) {
        float Jr[7];
        row_from_pix(P, fx, fy, x0b + c * HW_, x1b + c * HW_, Jr);
        const float wr = weights[b * CHW_ + c * HW_ + p] * Jr[6];
#pragma unroll
        for (int i = 0; i < 6; ++i) acc6[i] += Jr[i] * wr;
    }

    __shared__ float R6[6];
    if (tid < 6) R6[tid] = 0.f;
    __syncthreads();
#pragma unroll
    for (int i = 0; i < 6; ++i) atomicAdd(&R6[i], acc6[i]);
    __syncthreads();
    if (tid < 6) atomicAdd(&jtr[b * 8 + tid], R6[tid]);
}

// ----------------------------------------------------------------- k_solve ---
__global__ void k_solve(float* __restrict__ ws, float* __restrict__ out,
                        int use_gram_rhs, int write_T)
{
    const int b = threadIdx.x;
    if (b >= B_) return;
    float* Tb  = ws + WS_T   + b * 16;
    float* Gb  = ws + WS_G   + b * 64;
    float* jb  = ws + WS_JTR + b * 8;

    float M[6][7];
#pragma unroll
    for (int i = 0; i < 6; ++i) {
#pragma unroll
        for (int j = 0; j < 6; ++j)
            M[i][j] = Gb[i * 8 + j] + ((i == j) ? DAMP_ : 0.f);
        M[i][6] = use_gram_rhs ? Gb[i * 8 + 6] : jb[i];
    }

    // Gaussian elimination with partial pivoting
    for (int k = 0; k < 6; ++k) {
        int piv = k;
        float best = fabsf(M[k][k]);
        for (int i = k + 1; i < 6; ++i) {
            const float v = fabsf(M[i][k]);
            if (v > best) { best = v; piv = i; }
        }
        if (piv != k)
            for (int j = k; j < 7; ++j) {
                const float t = M[k][j]; M[k][j] = M[piv][j]; M[piv][j] = t;
            }
        const float inv = 1.f / M[k][k];
        for (int i = k + 1; i < 6; ++i) {
            const float f = M[i][k] * inv;
            for (int j = k; j < 7; ++j) M[i][j] -= f * M[k][j];
        }
    }
    float xi[6];
    for (int i = 5; i >= 0; --i) {
        float s = M[i][6];
        for (int j = i + 1; j < 6; ++j) s -= M[i][j] * xi[j];
        xi[i] = s / M[i][i];
    }

    float nxi[6];
#pragma unroll
    for (int i = 0; i < 6; ++i) nxi[i] = -xi[i];
    float E[16];
    exp_se3_dev(nxi, E);

    float Tn[16];
#pragma unroll
    for (int i = 0; i < 4; ++i)
#pragma unroll
        for (int j = 0; j < 4; ++j)
            Tn[i * 4 + j] = Tb[i * 4 + 0] * E[0 * 4 + j]
                          + Tb[i * 4 + 1] * E[1 * 4 + j]
                          + Tb[i * 4 + 2] * E[2 * 4 + j]
                          + Tb[i * 4 + 3] * E[3 * 4 + j];
#pragma unroll
    for (int k = 0; k < 16; ++k) Tb[k] = Tn[k];
    if (write_T)
#pragma unroll
        for (int k = 0; k < 16; ++k) out[b * 16 + k] = Tn[k];

    // reset JtR accumulator for next pass
#pragma unroll
    for (int i = 0; i < 8; ++i) jb[i] = 0.f;
}

// ----------------------------------------------------------------- launch ---
extern "C" void kernel_launch(void* const* d_in, const int* in_sizes, int n_in,
                              void* d_out, int out_size, void* d_ws, size_t ws_size,
                              hipStream_t stream) {
    (void)in_sizes; (void)n_in; (void)out_size; (void)ws_size;
    const float* pose  = (const float*)d_in[0];
    const float* x0    = (const float*)d_in[1];
    const float* x1    = (const float*)d_in[2];
    const float* invD0 = (const float*)d_in[3];
    const float* invD1 = (const float*)d_in[4];
    const float* Kp    = (const float*)d_in[5];
    float* out = (float*)d_out;        // [0,128)=T, [128,...)=weights
    float* ws  = (float*)d_ws;

    float* Tw      = ws + WS_T;
    float* G       = ws + WS_G;
    float* jtr     = ws + WS_JTR;
    float* weights = out + 128;

    const dim3 grid(B_ * BPB);

    k_init<<<1, 64, 0, stream>>>(pose, ws);
    k_gram<<<grid, THREADS, 0, stream>>>(x0, x1, invD0, invD1, Kp, Tw, G, weights);
    k_solve<<<1, 32, 0, stream>>>(ws, out, /*use_gram_rhs=*/1, /*write_T=*/0);  // -> T1
    k_jtr<<<grid, THREADS, 0, stream>>>(x0, x1, invD0, invD1, Kp, Tw, jtr, weights);
    k_solve<<<1, 32, 0, stream>>>(ws, out, 0, 0);                               // -> T2
    k_jtr<<<grid, THREADS, 0, stream>>>(x0, x1, invD0, invD1, Kp, Tw, jtr, weights);
    k_solve<<<1, 32, 0, stream>>>(ws, out, 0, 1);                               // -> T3 (out)
}